// RecModel4RGI_7773890805862
// MI455X (gfx1250) — compile-verified
//
#include <hip/hip_runtime.h>
#include <stdint.h>

#define D 64
#define THREADS 256
#define SCAN_ITEMS 8
#define SCAN_TILE (THREADS * SCAN_ITEMS)   // 2048 elements per scan block
#define STAGE_CAP 1024                     // max edges staged in LDS per block (8 KB)

typedef float v4f __attribute__((ext_vector_type(4)));

// ---------------------------------------------------------------------------
// init: cur = acc = concat(user_table, group_table)
// ---------------------------------------------------------------------------
__global__ __launch_bounds__(THREADS) void rg_init(
    const float* __restrict__ user, const float* __restrict__ grp,
    float* __restrict__ cur, float* __restrict__ acc,
    int userF4, int totF4)
{
  int idx = blockIdx.x * THREADS + threadIdx.x;
  if (idx >= totF4) return;
  v4f v = (idx < userF4) ? ((const v4f*)user)[idx]
                         : ((const v4f*)grp)[idx - userF4];
  ((v4f*)cur)[idx] = v;
  ((v4f*)acc)[idx] = v;
}

// ---------------------------------------------------------------------------
// CSR build step 1: zero per-row counters
// ---------------------------------------------------------------------------
__global__ __launch_bounds__(THREADS) void rg_zero_u32(uint32_t* __restrict__ p, int n)
{
  int i = blockIdx.x * THREADS + threadIdx.x;
  if (i < n) p[i] = 0u;
}

// ---------------------------------------------------------------------------
// CSR build step 2: histogram rows (hardware u32 atomics, L2-resident)
// ---------------------------------------------------------------------------
__global__ __launch_bounds__(THREADS) void rg_count(
    const int* __restrict__ rows, uint32_t* __restrict__ cnt, int E)
{
  int i = blockIdx.x * THREADS + threadIdx.x;
  if (i < E) atomicAdd(&cnt[rows[i]], 1u);
}

// ---------------------------------------------------------------------------
// CSR build step 3a: per-block exclusive scan (2048 elems/block)
// ---------------------------------------------------------------------------
__global__ __launch_bounds__(THREADS) void rg_scan1(
    const uint32_t* __restrict__ cnt, uint32_t* __restrict__ rp,
    uint32_t* __restrict__ bsum, int n)
{
  __shared__ uint32_t s_sum[THREADS];
  __shared__ uint32_t s_off[THREADS];
  const int t = threadIdx.x;
  const long base = (long)blockIdx.x * SCAN_TILE + (long)t * SCAN_ITEMS;
  uint32_t v[SCAN_ITEMS];
  uint32_t run = 0;
#pragma unroll
  for (int k = 0; k < SCAN_ITEMS; ++k) {
    long idx = base + k;
    v[k] = run;                                   // exclusive
    run += (idx < n) ? cnt[idx] : 0u;
  }
  s_sum[t] = run;
  __syncthreads();
  if (t == 0) {
    uint32_t a = 0;
    for (int i = 0; i < THREADS; ++i) { s_off[i] = a; a += s_sum[i]; }
    bsum[blockIdx.x] = a;
  }
  __syncthreads();
  uint32_t off = s_off[t];
#pragma unroll
  for (int k = 0; k < SCAN_ITEMS; ++k) {
    long idx = base + k;
    if (idx < n) rp[idx] = v[k] + off;
  }
}

// ---------------------------------------------------------------------------
// CSR build step 3b: scan the block sums; also publish rp[n] = E
// ---------------------------------------------------------------------------
__global__ void rg_scan2(uint32_t* __restrict__ bsum, uint32_t* __restrict__ rp,
                         int nb, int n)
{
  if (threadIdx.x == 0 && blockIdx.x == 0) {
    uint32_t a = 0;
    for (int i = 0; i < nb; ++i) { uint32_t s = bsum[i]; bsum[i] = a; a += s; }
    rp[n] = a;                                    // == E
  }
}

// ---------------------------------------------------------------------------
// CSR build step 3c: add block offsets; mirror into the scatter cursor
// ---------------------------------------------------------------------------
__global__ __launch_bounds__(THREADS) void rg_scan3(
    uint32_t* __restrict__ rp, uint32_t* __restrict__ cursor,
    const uint32_t* __restrict__ bsum, int n)
{
  const int t = threadIdx.x;
  const long base = (long)blockIdx.x * SCAN_TILE + t;
  const uint32_t off = bsum[blockIdx.x];
#pragma unroll
  for (int k = 0; k < SCAN_ITEMS; ++k) {
    long idx = base + (long)k * THREADS;
    if (idx < n) {
      uint32_t v = rp[idx] + off;
      rp[idx]     = v;
      cursor[idx] = v;
    }
  }
}

// ---------------------------------------------------------------------------
// CSR build step 4: scatter packed (col, val) records into row-sorted order
// ---------------------------------------------------------------------------
__global__ __launch_bounds__(THREADS) void rg_scatter(
    const int* __restrict__ rows, const int* __restrict__ cols,
    const float* __restrict__ vals, uint32_t* __restrict__ cursor,
    int2* __restrict__ scv, int E)
{
  int i = blockIdx.x * THREADS + threadIdx.x;
  if (i < E) {
    uint32_t pos = atomicAdd(&cursor[rows[i]], 1u);
    scv[pos] = make_int2(cols[i], __float_as_int(vals[i]));
  }
}

// ---------------------------------------------------------------------------
// CSR SpMM layer: nxt[r] = sum_j val*cur[col]; acc[r] += nxt[r]. No atomics.
// Block = 16 rows; the block's edges are one contiguous segment (sorted by
// row), async-staged into LDS with CDNA5 global->LDS b64 loads.
// 16 lanes per row, float4 per lane -> 256B coalesced row transactions.
// cur gathers use default (RT) policy to stay L2-resident; single-touch
// acc/nxt traffic uses non-temporal policy to protect L2 capacity.
// ---------------------------------------------------------------------------
__global__ __launch_bounds__(THREADS) void rg_csr_spmm(
    const uint32_t* __restrict__ rp, const int2* __restrict__ scv,
    const float* __restrict__ cur, float* __restrict__ nxt,
    float* __restrict__ acc, int n)
{
  __shared__ int2 s_cv[STAGE_CAP];
  const int t = threadIdx.x;
  const int rbase = blockIdx.x * 16;
  int rend = rbase + 16; if (rend > n) rend = n;

  const uint32_t blkS = rp[rbase];
  const uint32_t blkE = rp[rend];
  const uint32_t len  = blkE - blkS;
  const bool staged = (len <= (uint32_t)STAGE_CAP);

  if (staged) {
    uint32_t lbase = (uint32_t)(uintptr_t)(&s_cv[0]);
    for (uint32_t k = t; k < len; k += THREADS) {
      uint64_t g = (uint64_t)(uintptr_t)(scv + blkS + k);
      uint32_t l = lbase + k * 8u;
      asm volatile("global_load_async_to_lds_b64 %0, %1, off" :: "v"(l), "v"(g) : "memory");
    }
    asm volatile("s_wait_asynccnt 0" ::: "memory");
  }
  __syncthreads();

  const int lane4 = (t & 15) * 4;
  const int r = rbase + (t >> 4);
  if (r < n) {
    const uint32_t s = rp[r], e = rp[r + 1];
    v4f a = (v4f){0.f, 0.f, 0.f, 0.f};
    if (staged) {
#pragma unroll 2
      for (uint32_t j = s; j < e; ++j) {
        int2 cv = s_cv[j - blkS];                 // ds broadcast, 16 lanes
        float v = __int_as_float(cv.y);
        const v4f x = *(const v4f*)(cur + (size_t)cv.x * D + lane4);
        a += x * v;
      }
    } else {                                      // rare huge segment
#pragma unroll 2
      for (uint32_t j = s; j < e; ++j) {
        int2 cv = scv[j];
        float v = __int_as_float(cv.y);
        const v4f x = *(const v4f*)(cur + (size_t)cv.x * D + lane4);
        a += x * v;
      }
    }
    // exactly-once row write: non-temporal to keep 'cur' resident in L2
    __builtin_nontemporal_store(a, (v4f*)(nxt + (size_t)r * D + lane4));
    v4f* ap = (v4f*)(acc + (size_t)r * D + lane4);
    v4f av = __builtin_nontemporal_load(ap);
    __builtin_nontemporal_store(av + a, ap);
  }
}

// ---------------------------------------------------------------------------
// gather: six (B,64) output blocks; /4 average folded into the first three
// ---------------------------------------------------------------------------
__global__ __launch_bounds__(THREADS) void rg_gather(
    const float* __restrict__ acc, const float* __restrict__ user,
    const float* __restrict__ grp, const int* __restrict__ ui,
    const int* __restrict__ pg, const int* __restrict__ ng,
    float* __restrict__ out, int NU, int B)
{
  int t = threadIdx.x;
  int d = t & 63;
  int b = blockIdx.x * 4 + (t >> 6);
  if (b >= B) return;
  int u = ui[b], p = pg[b], n = ng[b];
  size_t BD = (size_t)B * D;
  size_t bd = (size_t)b * D + d;
  out[0 * BD + bd] = acc[(size_t)u * D + d]        * 0.25f;
  out[1 * BD + bd] = acc[(size_t)(NU + p) * D + d] * 0.25f;
  out[2 * BD + bd] = acc[(size_t)(NU + n) * D + d] * 0.25f;
  out[3 * BD + bd] = user[(size_t)u * D + d];
  out[4 * BD + bd] = grp[(size_t)p * D + d];
  out[5 * BD + bd] = grp[(size_t)n * D + d];
}

// ---------------------------------------------------------------------------
extern "C" void kernel_launch(void* const* d_in, const int* in_sizes, int n_in,
                              void* d_out, int out_size, void* d_ws, size_t ws_size,
                              hipStream_t stream)
{
  const float* user = (const float*)d_in[0];
  const float* grp  = (const float*)d_in[1];
  const float* vals = (const float*)d_in[2];
  const int*   rows = (const int*)d_in[3];
  const int*   cols = (const int*)d_in[4];
  const int*   ui   = (const int*)d_in[5];
  const int*   pg   = (const int*)d_in[6];
  const int*   ng   = (const int*)d_in[7];
  float* out = (float*)d_out;

  const int NU   = in_sizes[0] / D;     // 200000
  const int NG   = in_sizes[1] / D;     // 50000
  const int Ntot = NU + NG;             // 250000
  const int E    = in_sizes[3];         // 4000000
  const int B    = in_sizes[5];         // 8192

  // ---- workspace carve-up (sizes in elements; all regions 1KB-aligned) ----
  const size_t NF   = (size_t)Ntot * D;              // 16M floats per table
  const size_t NPAD = ((size_t)Ntot + 1024) & ~1023ull;
  char* w = (char*)d_ws;
  float*    cur    = (float*)w;            w += NF * sizeof(float);
  float*    nxt    = (float*)w;            w += NF * sizeof(float);
  float*    acc    = (float*)w;            w += NF * sizeof(float);
  uint32_t* rp     = (uint32_t*)w;         w += NPAD * sizeof(uint32_t);   // N+1 used
  uint32_t* cursor = (uint32_t*)w;         w += NPAD * sizeof(uint32_t);
  uint32_t* cnt    = (uint32_t*)w;         w += NPAD * sizeof(uint32_t);
  uint32_t* bsum   = (uint32_t*)w;         w += 1024 * sizeof(uint32_t);
  int2*     scv    = (int2*)w;             // E records (32 MB)

  const int totF4  = (int)(NF / 4);
  const int userF4 = NU * D / 4;
  const int gridN  = (totF4 + THREADS - 1) / THREADS;       // 15625
  const int gridE  = (E + THREADS - 1) / THREADS;           // 15625
  const int gridZ  = (Ntot + THREADS - 1) / THREADS;        // 977
  const int NB     = (Ntot + SCAN_TILE - 1) / SCAN_TILE;    // 123
  const int gridR  = (Ntot + 15) / 16;                      // 15625

  // embeddings
  rg_init<<<gridN, THREADS, 0, stream>>>(user, grp, cur, acc, userF4, totF4);

  // CSR build (counting sort by row)
  rg_zero_u32<<<gridZ, THREADS, 0, stream>>>(cnt, Ntot);
  rg_count   <<<gridE, THREADS, 0, stream>>>(rows, cnt, E);
  rg_scan1   <<<NB,    THREADS, 0, stream>>>(cnt, rp, bsum, Ntot);
  rg_scan2   <<<1, 1, 0, stream>>>(bsum, rp, NB, Ntot);
  rg_scan3   <<<NB,    THREADS, 0, stream>>>(rp, cursor, bsum, Ntot);
  rg_scatter <<<gridE, THREADS, 0, stream>>>(rows, cols, vals, cursor, scv, E);

  // 3 atomic-free propagation layers with fused accumulate
  for (int l = 0; l < 3; ++l) {
    rg_csr_spmm<<<gridR, THREADS, 0, stream>>>(rp, scv, cur, nxt, acc, Ntot);
    float* tmp = cur; cur = nxt; nxt = tmp;
  }

  // outputs
  rg_gather<<<(B + 3) / 4, THREADS, 0, stream>>>(acc, user, grp, ui, pg, ng, out, NU, B);
}